// EquivariantProductBasisBlock_3547642987261
// MI455X (gfx1250) — compile-verified
//
#include <hip/hip_runtime.h>

typedef __attribute__((ext_vector_type(2))) float v2f;
typedef __attribute__((ext_vector_type(8))) float v8f;
typedef unsigned int v4u __attribute__((ext_vector_type(4)));
typedef int v8i_ __attribute__((ext_vector_type(8)));
typedef int v4i_ __attribute__((ext_vector_type(4)));

#define N_NODES_C 512
#define CH      128
#define NELL    16
#define NP3     23
#define K3      (NELL * NP3)   // 368
#define NSTEP   (K3 / 4)       // 92 WMMA f32 16x16x4 steps
#define TPB     256
#define TILE_F  (16 * K3)      // 5888 floats per A tile

#if __has_builtin(__builtin_amdgcn_tensor_load_to_lds) && \
    __has_builtin(__builtin_amdgcn_s_wait_tensorcnt)
#define USE_TDM 1
#endif

struct Params {
  const float* x;      // [512,128,16]
  const float* sc;     // [512,2048]
  const float* attrs;  // [512,10]
  const float* U1[4];  // (m,16,1)
  const float* U2[4];  // (m,16,16,4)
  const float* U3[4];  // (m,16,16,16,23)
  const float* W1[4];  // (10,1,128)
  const float* W2[4];  // (10,4,128)
  const float* W3[4];  // (10,23,128)
  const float* linw[4];// (128,128)
  float* out;          // [512,2048]
};

#ifdef USE_TDM
// 1-D TDM DMA: copy TILE_F f32 from gsrc into LDS at lds_dst.
// D# layout per CDNA5 ISA 8.3/8.4: group0={count=1, lds_addr, global_addr, type=2},
// group1={data_size=4B, tensor_dim0=tile_dim0=5888, 1-D tile}, groups2/3 unused.
// This toolchain exposes the 6-arg builtin:
//   (uint32x4 g0, int32x8 g1, int32x4 g2, int32x4 g3, int32x8 g4, i32 cpol)
__device__ __forceinline__ void tdm_load_tile(const float* gsrc, void* lds_dst) {
  const unsigned long long ga = (unsigned long long)(size_t)gsrc;
  const unsigned lds = (unsigned)(size_t)lds_dst;  // low 32 bits = LDS byte offset
  v4u g0;
  g0.x = 1u;                                            // count=1 (valid user D#)
  g0.y = lds;                                           // lds_addr
  g0.z = (unsigned)ga;                                  // global_addr[31:0]
  g0.w = (unsigned)((ga >> 32) & 0x01FFFFFFull)         // global_addr[56:32]
         | 0x80000000u;                                 // type=2 ("image")
  v8i_ g1;
  g1[0] = 0x00020000;                                   // data_size=2 (4B), no flags
  g1[1] = (int)((TILE_F & 0xFFFFu) << 16);              // tensor_dim0[15:0]
  g1[2] = (int)((TILE_F >> 16) | (1u << 16));           // tensor_dim0[31:16] | tensor_dim1=1
  g1[3] = (int)((unsigned)TILE_F << 16);                // tile_dim0 = 5888
  g1[4] = 0;                                            // tile_dim1=0, tile_dim2=0 (1-D)
  g1[5] = TILE_F;                                       // tensor_dim0_stride lo32
  g1[6] = 0;                                            // stride hi | dim1_stride lo
  g1[7] = 0;
  v4i_ z4 = {0, 0, 0, 0};
  v8i_ z8 = {0, 0, 0, 0, 0, 0, 0, 0};
  __builtin_amdgcn_tensor_load_to_lds(g0, g1, z4, z4, z8, 0);
}
#endif

__global__ __launch_bounds__(TPB) void mace_epb_kernel(Params prm) {
  const int  b    = blockIdx.x;
  const int  tid  = threadIdx.x;
  const int  lane = tid & 31;
  const int  wid  = tid >> 5;
  const int  colL = lane & 15;           // column inside the wave's 16-col tile
  const int  col  = (wid << 4) + colL;   // channel 0..127 of this node
  const bool hi   = lane >= 16;

  __shared__ __align__(16) float xbuf [CH * NELL];    // x[c][i]
  __shared__ __align__(16) float abuf [2][TILE_F];    // double-buffered A tile [q][kk]
  __shared__ __align__(16) float w3buf[NP3 * CH];     // [k][c]
  __shared__ __align__(16) float w2buf[4 * CH];       // [k][c]
  __shared__ __align__(16) float w1buf[CH];
  __shared__ __align__(16) float u2buf[16 * 16 * 4];  // [p][q][k]
  __shared__ __align__(16) float u1buf[16];           // [p]
  __shared__ __align__(16) float zbuf [CH * 16];      // z[c][Lm] pre-linear
  __shared__ float attr_s[10];

#ifdef USE_TDM
  // Kick off DMA of the very first A tile (L=0,m=0,p=0) immediately.
  if (wid == 0) tdm_load_tile(prm.U3[0], &abuf[0][0]);
  int par = 0;
#endif

  // ---- stage node data -------------------------------------------------
  const float* xg = prm.x + (size_t)b * (CH * NELL);
  for (int j = tid; j < CH * NELL; j += TPB) xbuf[j] = xg[j];
  if (tid < 10) attr_s[tid] = prm.attrs[b * 10 + tid];
  __syncthreads();

  float xr[NELL];                       // x[col][0..15] in registers
#pragma unroll
  for (int i = 0; i < NELL; ++i) xr[i] = xbuf[(col << 4) + i];
  float xqr[8];                         // x values for this lane's 8 q-rows
#pragma unroll
  for (int v = 0; v < 8; ++v) xqr[v] = hi ? xr[v + 8] : xr[v];

  int Lm = 0;
  for (int L = 0; L < 4; ++L) {
    const int mcount = 2 * L + 1;
    __syncthreads();                    // previous readers of w*buf are done
    {
      const float* W3p = prm.W3[L];
      for (int j = tid; j < NP3 * CH; j += TPB) {
        float s = 0.f;
#pragma unroll
        for (int e = 0; e < 10; ++e) s += attr_s[e] * W3p[e * NP3 * CH + j];
        w3buf[j] = s;
      }
      const float* W2p = prm.W2[L];
      for (int j = tid; j < 4 * CH; j += TPB) {
        float s = 0.f;
#pragma unroll
        for (int e = 0; e < 10; ++e) s += attr_s[e] * W2p[e * 4 * CH + j];
        w2buf[j] = s;
      }
      if (tid < CH) {
        const float* W1p = prm.W1[L];
        float s = 0.f;
#pragma unroll
        for (int e = 0; e < 10; ++e) s += attr_s[e] * W1p[e * CH + tid];
        w1buf[tid] = s;
      }
    }
    __syncthreads();

    float w3r[NP3];
#pragma unroll
    for (int k = 0; k < NP3; ++k) w3r[k] = w3buf[k * CH + col];
    float w2r[4];
#pragma unroll
    for (int k = 0; k < 4; ++k) w2r[k] = w2buf[k * CH + col];
    const float w1r = w1buf[col];

    for (int m = 0; m < mcount; ++m, ++Lm) {
      __syncthreads();                  // previous u2buf readers are done
      {
        const float* U2p = prm.U2[L] + m * (16 * 16 * 4);
        for (int j = tid; j < 16 * 16 * 4; j += TPB) u2buf[j] = U2p[j];
        if (tid < 16) u1buf[tid] = prm.U1[L][m * 16 + tid];
      }
      float accz = 0.f;                 // per-lane half of out1[col,(L,m)]
      for (int pp = 0; pp < 16; ++pp) {
        const float* blk = prm.U3[L] + (size_t)((m * 16 + pp) * 16) * K3;
#ifdef USE_TDM
        // Current tile (issued one iteration ago) must have landed; the same
        // barrier also retires all readers of the buffer we refill next.
        if (wid == 0) __builtin_amdgcn_s_wait_tensorcnt(0);
        __syncthreads();
        {
          const float* nblk = nullptr;
          if (pp < 15) {
            nblk = blk + TILE_F;
          } else {
            int nm = m + 1, nL = L;
            if (nm == mcount) { nm = 0; nL = L + 1; }
            if (nL < 4) nblk = prm.U3[nL] + (size_t)(nm * 256) * K3;
          }
          if (wid == 0 && nblk) tdm_load_tile(nblk, &abuf[par ^ 1][0]);
        }
        const float* at = &abuf[par][0];
        par ^= 1;
#else
        __syncthreads();                // previous abuf readers done
        for (int j = tid; j < TILE_F; j += TPB) abuf[0][j] = blk[j];
        if (pp < 15) __builtin_prefetch(blk + TILE_F + tid * 23, 0, 3);
        __syncthreads();
        const float* at = &abuf[0][0];
#endif

        // C[q=16][col=16] = A[16 x 368] * Z[368 x 16],  Z[(i,k),col]=x[i]*w3[k]
        v8f C = {0.f, 0.f, 0.f, 0.f, 0.f, 0.f, 0.f, 0.f};
        const int arow = colL * K3 + (hi ? 2 : 0);
#pragma unroll
        for (int s = 0; s < NSTEP; ++s) {
          const int kk0 = 4 * s;
          v2f a = *(const v2f*)&at[arow + kk0];
          const float p0 = xr[(kk0    ) / NP3] * w3r[(kk0    ) % NP3];
          const float p1 = xr[(kk0 + 1) / NP3] * w3r[(kk0 + 1) % NP3];
          const float p2 = xr[(kk0 + 2) / NP3] * w3r[(kk0 + 2) % NP3];
          const float p3 = xr[(kk0 + 3) / NP3] * w3r[(kk0 + 3) % NP3];
          v2f bf;
          bf.x = hi ? p2 : p0;
          bf.y = hi ? p3 : p1;
          C = __builtin_amdgcn_wmma_f32_16x16x4_f32(
              false, a, false, bf, (short)0, C, false, false);
        }

        // epilogue: c2 = out3 + U2*w2 ; out2 = sum_q c2*x[q] ; fold over p
        const float xp = xbuf[(col << 4) + pp];
        float t = 0.f;
#pragma unroll
        for (int v = 0; v < 8; ++v) {
          const int q = (hi ? 8 : 0) + v;
          const float4 u2 = *(const float4*)&u2buf[(pp << 6) + (q << 2)];
          const float c2 = C[v] + u2.x * w2r[0] + u2.y * w2r[1] +
                                  u2.z * w2r[2] + u2.w * w2r[3];
          t += c2 * xqr[v];
        }
        accz += (t + 0.5f * u1buf[pp] * w1r) * xp;  // 0.5: both lane halves add it
      }
      const float tot = accz + __shfl_xor(accz, 16, 32);
      if (!hi) zbuf[col * 16 + Lm] = tot;
    }
  }
  __syncthreads();

  // ---- fused equivariant linear (per-irrep channel mix) + sc -----------
  {
    const int d = tid & 127;
    const int g = tid >> 7;
    const float invs = 0.08838834764831845f;  // 1/sqrt(128)
#pragma unroll
    for (int j = 0; j < 8; ++j) {
      const int lm = g * 8 + j;
      int L, mm, off;
      if (lm == 0)     { L = 0; mm = 0;      off = 0;    }
      else if (lm < 4) { L = 1; mm = lm - 1; off = 128;  }
      else if (lm < 9) { L = 2; mm = lm - 4; off = 512;  }
      else             { L = 3; mm = lm - 9; off = 1152; }
      const float* lw = prm.linw[L];
      float acc = 0.f;
      for (int c = 0; c < CH; ++c) acc += zbuf[c * 16 + lm] * lw[c * CH + d];
      const int mlen = 2 * L + 1;
      const size_t oidx = (size_t)b * 2048 + off + d * mlen + mm;
      prm.out[oidx] = acc * invs + prm.sc[oidx];
    }
  }
}

extern "C" void kernel_launch(void* const* d_in, const int* in_sizes, int n_in,
                              void* d_out, int out_size, void* d_ws, size_t ws_size,
                              hipStream_t stream) {
  (void)in_sizes; (void)n_in; (void)out_size; (void)d_ws; (void)ws_size;
  Params prm;
  prm.x     = (const float*)d_in[0];
  prm.sc    = (const float*)d_in[1];
  prm.attrs = (const float*)d_in[2];
  for (int L = 0; L < 4; ++L) {
    prm.U1[L]   = (const float*)d_in[3 + 3 * L + 0];
    prm.U2[L]   = (const float*)d_in[3 + 3 * L + 1];
    prm.U3[L]   = (const float*)d_in[3 + 3 * L + 2];
    prm.W1[L]   = (const float*)d_in[15 + 3 * L + 0];
    prm.W2[L]   = (const float*)d_in[15 + 3 * L + 1];
    prm.W3[L]   = (const float*)d_in[15 + 3 * L + 2];
    prm.linw[L] = (const float*)d_in[27 + L];
  }
  prm.out = (float*)d_out;
  mace_epb_kernel<<<dim3(N_NODES_C), dim3(TPB), 0, stream>>>(prm);
}